// SPPostProcessingBlock_91096256348433
// MI455X (gfx1250) — compile-verified
//
#include <hip/hip_runtime.h>

#define THREADS 256   // 8 wave32 per block

// out[i] = adj[i] > 0.5f ? 1.0f : 0.0f   (closed form of the 1000-step scan)
//
// Streaming path: per-lane 128-bit async copies global->LDS, double buffered
// with s_wait_asynccnt (ASYNCcnt completes in order), then ds_load_b128 +
// compare + global_store_b128. Pure HBM-bandwidth-bound (~16.8 MB total).
__global__ __launch_bounds__(THREADS)
void sp_postproc_stream(const float* __restrict__ in, float* __restrict__ out,
                        int n4, int n, int tilesPerBlock) {
  __shared__ float4 stage[2][THREADS];
  const int tid = (int)threadIdx.x;
  const long long base = (long long)blockIdx.x * tilesPerBlock * THREADS;

  const float4* __restrict__ in4 = (const float4*)in;
  float4* __restrict__ out4 = (float4*)out;

  // Issue one 16B-per-lane async global->LDS copy for tile t.
  auto issue = [&](int t) {
    long long idx = base + (long long)t * THREADS + tid;
    if (idx >= n4) idx = 0;  // clamped dummy load; store is guarded below
    unsigned ldsAddr = (unsigned)(size_t)(&stage[t & 1][tid]);
    unsigned long long gaddr =
        (unsigned long long)(size_t)(in4 + idx);
    asm volatile("global_load_async_to_lds_b128 %0, %1, off"
                 :
                 : "v"(ldsAddr), "v"(gaddr)
                 : "memory");
  };

  issue(0);
  for (int t = 0; t < tilesPerBlock; ++t) {
    if (t + 1 < tilesPerBlock) {
      issue(t + 1);
      // 2 outstanding; async loads complete in order, so <=1 means tile t done.
      asm volatile("s_wait_asynccnt 1" ::: "memory");
    } else {
      asm volatile("s_wait_asynccnt 0" ::: "memory");
    }

    float4 v = stage[t & 1][tid];
    float4 r;
    r.x = v.x > 0.5f ? 1.0f : 0.0f;
    r.y = v.y > 0.5f ? 1.0f : 0.0f;
    r.z = v.z > 0.5f ? 1.0f : 0.0f;
    r.w = v.w > 0.5f ? 1.0f : 0.0f;

    long long idx = base + (long long)t * THREADS + tid;
    if (idx < n4) out4[idx] = r;
  }

  // Scalar tail for n not divisible by 4 (not hit for the 2M-element shape).
  if (blockIdx.x == 0) {
    int tail = n - n4 * 4;
    if (tid < tail) {
      int i = n4 * 4 + tid;
      out[i] = in[i] > 0.5f ? 1.0f : 0.0f;
    }
  }
}

extern "C" void kernel_launch(void* const* d_in, const int* in_sizes, int n_in,
                              void* d_out, int out_size, void* d_ws, size_t ws_size,
                              hipStream_t stream) {
  (void)n_in; (void)d_ws; (void)ws_size;
  const float* in = (const float*)d_in[0];
  float* out = (float*)d_out;

  int n = out_size;            // 2*1024*1024 for the reference shape
  if (n <= 0) n = in_sizes[0];
  int n4 = n / 4;

  const int tilesPerBlock = 4; // 4 tiles x 256 lanes x 16B = 16 KB per block
  long long work = (long long)THREADS * tilesPerBlock;
  int blocks = (int)((n4 + work - 1) / work);
  if (blocks < 1) blocks = 1;

  sp_postproc_stream<<<blocks, THREADS, 0, stream>>>(in, out, n4, n, tilesPerBlock);
}